// SegmantationLoss_33947421507995
// MI455X (gfx1250) — compile-verified
//
#include <hip/hip_runtime.h>
#include <hip/hip_bf16.h>
#include <math.h>

// ---------------------------------------------------------------------------
// SegmantationLoss for MI455X (gfx1250, wave32).
//  k_reduce : masked einsum reductions via V_WMMA_F32_16X16X4_F32 (exact f32)
//  k_gauss  : gaussians, error->u32 sort keys (label in LSB), seed fg term
//  radix    : 4-pass LSD sort, 32 segments of 512K keys, ballot multi-split
//  k_lovasz : streaming label prefix-scan -> lovasz hinge per segment
//  k_final  : combine lovasz + sigma + seed losses -> scalar
// Workspace needed: ~139 MB.
// ---------------------------------------------------------------------------

typedef float v2f __attribute__((ext_vector_type(2)));
typedef float v8f __attribute__((ext_vector_type(8)));

#define B_    4
#define H_    512
#define W_    1024
#define P_    (H_ * W_)        // 524288
#define NI    8
#define NSEG  (NI * B_)        // 32
#define RB    128              // radix blocks per segment
#define RBI   4096             // items per radix block
#define MESH_INV (1.0f / 1024.0f)

// workspace byte offsets
#define OFF_SUMS  0                                   // B*256 f32   (4 KB)
#define OFF_STAT  4096                                // NSEG*8 f32
#define OFF_BG    5120                                // B f32
#define OFF_FG    5184                                // NSEG f32
#define OFF_LOV   5312                                // NSEG f32
#define OFF_CNTS  8192                                // NSEG*256*RB u32 (4 MB)
#define OFF_ITEMA (OFF_CNTS + (size_t)NSEG * 256 * RB * 4)
#define OFF_ITEMB (OFF_ITEMA + (size_t)NSEG * P_ * 4)
#define WS_NEEDED (OFF_ITEMB + (size_t)NSEG * P_ * 4)

// ---------------------------------------------------------------------------
__global__ void k_zero(float* __restrict__ sums, float* __restrict__ fg) {
  int i = blockIdx.x * blockDim.x + threadIdx.x;
  if (i < B_ * 256) sums[i] = 0.f;
  if (i < NSEG)     fg[i]   = 0.f;
}

// ---------------------------------------------------------------------------
// Pass 1: D[m][n] += sum_p mask(m,p) * feat(n,p) using f32 WMMA (16x16x4).
// Rows m=0..7: instance masks (t==m+1); m=8: background (t==0).
// Cols n: 0=emb_y 1=emb_x 2=pps_y 3=pps_x 4=1 5=pps_y^2 6=pps_x^2 7=seed^2
__global__ __launch_bounds__(256) void k_reduce(
    const float* __restrict__ xvec, const float* __restrict__ xsig,
    const float* __restrict__ xseed, const int* __restrict__ t,
    float* __restrict__ sums)
{
  __shared__ float feats[8][32][16];
  __shared__ int   tc[8][32];
  __shared__ float red[256 * 8];

  const int tid  = threadIdx.x;
  const int wave = tid >> 5, lane = tid & 31;
  const int b    = blockIdx.y;

  // zero padded feature columns 8..15 once
  for (int q = tid; q < 8 * 32; q += 256) {
    int w = q >> 5, l = q & 31;
    #pragma unroll
    for (int c = 8; c < 16; ++c) feats[w][l][c] = 0.f;
  }
  __syncthreads();

  v8f acc = {0.f, 0.f, 0.f, 0.f, 0.f, 0.f, 0.f, 0.f};

  const int pixBase = blockIdx.x * 2048;
  const float* vy = xvec  + (size_t)b * 2 * P_;
  const float* vx = vy + P_;
  const float* sy = xsig  + (size_t)b * 2 * P_;
  const float* sx = sy + P_;
  const float* sd = xseed + (size_t)b * P_;
  const int*   tt = t     + (size_t)b * P_;

  const int mn = lane & 15;

  for (int it = 0; it < 8; ++it) {
    int p = pixBase + it * 256 + wave * 32 + lane;
    float fvy = vy[p], fvx = vx[p];
    float fsy = sy[p], fsx = sx[p];
    float fse = sd[p];
    int   ti  = tt[p];

    float ey = tanhf(fvy) + (float)(p >> 10)       * MESH_INV;
    float ex = tanhf(fvx) + (float)(p & (W_ - 1))  * MESH_INV;
    float sg = 1.f / (1.f + expf(-fse));

    feats[wave][lane][0] = ey;
    feats[wave][lane][1] = ex;
    feats[wave][lane][2] = fsy;
    feats[wave][lane][3] = fsx;
    feats[wave][lane][4] = 1.f;
    feats[wave][lane][5] = fsy * fsy;
    feats[wave][lane][6] = fsx * fsx;
    feats[wave][lane][7] = sg * sg;
    tc[wave][lane] = ti;
    __syncthreads();

    #pragma unroll
    for (int j = 0; j < 8; ++j) {
      // A 16x4 f32: VGPR0 = K0 (lanes 0-15) / K2 (lanes 16-31); VGPR1 = K1/K3
      int k0 = 4 * j + ((lane < 16) ? 0 : 2);
      int k1 = k0 + 1;
      int t0 = tc[wave][k0], t1 = tc[wave][k1];
      v2f a, bm;
      a.x = (mn < 8) ? ((t0 == mn + 1) ? 1.f : 0.f)
                     : ((mn == 8) ? ((t0 == 0) ? 1.f : 0.f) : 0.f);
      a.y = (mn < 8) ? ((t1 == mn + 1) ? 1.f : 0.f)
                     : ((mn == 8) ? ((t1 == 0) ? 1.f : 0.f) : 0.f);
      bm.x = feats[wave][k0][mn];
      bm.y = feats[wave][k1][mn];
      acc = __builtin_amdgcn_wmma_f32_16x16x4_f32(
          false, a, false, bm, (short)0, acc, false, false);
    }
    __syncthreads();
  }

  // C/D layout: VGPR r -> M = r + 8*(lane>=16), N = lane&15
  const int nn  = lane & 15;
  const int mhi = (lane >> 4) * 8;
  #pragma unroll
  for (int r = 0; r < 8; ++r) {
    int m = r + mhi;
    red[(m * 16 + nn) * 8 + wave] = acc[r];
  }
  __syncthreads();

  float s = 0.f;
  #pragma unroll
  for (int w = 0; w < 8; ++w) s += red[tid * 8 + w];
  atomicAdd(&sums[b * 256 + tid], s);
}

// ---------------------------------------------------------------------------
__global__ void k_stats(const float* __restrict__ sums,
                        float* __restrict__ stat, float* __restrict__ bgl)
{
  int s = threadIdx.x;
  if (s < NSEG) {
    int n = s / B_, b = s % B_;
    const float* r = sums + b * 256 + n * 16;
    float cnt  = r[4];
    float safe = fmaxf(cnt, 1.f);
    float pres = (cnt > 0.f) ? 1.f : 0.f;
    float cy = r[0] / safe, cx = r[1] / safe;
    float my = r[2] / safe, mx = r[3] / safe;
    float psy = expf(10.f * my);
    float psx = expf(10.f * mx);
    float sl = (r[5] - 2.f * my * r[2] + my * my * cnt +
                r[6] - 2.f * mx * r[3] + mx * mx * cnt) / safe;
    float* o = stat + s * 8;
    o[0] = cnt; o[1] = safe; o[2] = pres; o[3] = cy;
    o[4] = cx;  o[5] = psy;  o[6] = psx;  o[7] = sl;
  }
  if (s < B_) {
    const float* r = sums + s * 256 + 8 * 16;   // background row
    bgl[s] = r[7] / fmaxf(r[4], 1.f);
  }
}

// ---------------------------------------------------------------------------
// error -> descending-order u32 key, label packed into LSB
__device__ __forceinline__ unsigned err_to_key(float err, int lab) {
  unsigned fb  = __float_as_uint(err);
  unsigned asc = (fb & 0x80000000u) ? ~fb : (fb ^ 0x80000000u);
  return ((~asc) & ~1u) | (unsigned)lab;
}
__device__ __forceinline__ float key_to_err(unsigned key) {
  unsigned asc = ~key;
  unsigned fb  = (asc & 0x80000000u) ? (asc ^ 0x80000000u) : ~asc;
  return __uint_as_float(fb);
}

__global__ __launch_bounds__(256) void k_gauss(
    const float* __restrict__ xvec, const float* __restrict__ xseed,
    const int* __restrict__ t, const float* __restrict__ stat,
    float* __restrict__ fg, unsigned* __restrict__ items)
{
  __shared__ float fgacc[NI];
  const int tid = threadIdx.x;
  const int b   = blockIdx.y;
  if (tid < NI) fgacc[tid] = 0.f;
  __syncthreads();

  float cy[NI], cx[NI], psy[NI], psx[NI];
  #pragma unroll
  for (int n = 0; n < NI; ++n) {
    const float* o = stat + (n * B_ + b) * 8;
    cy[n] = o[3]; cx[n] = o[4]; psy[n] = o[5]; psx[n] = o[6];
  }

  const float* vy  = xvec  + (size_t)b * 2 * P_;
  const float* vx  = vy + P_;
  const float* sdp = xseed + (size_t)b * P_;
  const int*   tt  = t     + (size_t)b * P_;

  int base = blockIdx.x * 2048;
  for (int it = 0; it < 8; ++it) {
    int p = base + it * 256 + tid;
    float ey = tanhf(vy[p]) + (float)(p >> 10)      * MESH_INV;
    float ex = tanhf(vx[p]) + (float)(p & (W_ - 1)) * MESH_INV;
    float sg = 1.f / (1.f + expf(-sdp[p]));
    int   ti = tt[p];
    float gsel = 0.f;
    #pragma unroll
    for (int n = 0; n < NI; ++n) {
      float dy = ey - cy[n], dx = ex - cx[n];
      float dist = dy * dy * psy[n] + dx * dx * psx[n];
      float g = expf(-0.5f * dist);
      int lab = (ti == n + 1) ? 1 : 0;
      float sfl = lab ? 1.f : -1.f;
      float err = 1.f - (2.f * g - 1.f) * sfl;
      items[(size_t)(n * B_ + b) * P_ + p] = err_to_key(err, lab);
      if (lab) gsel = g;
    }
    if (ti >= 1 && ti <= NI) {
      float d = sg - gsel;
      atomicAdd(&fgacc[ti - 1], d * d);
    }
  }
  __syncthreads();
  if (tid < NI) atomicAdd(&fg[tid * B_ + b], fgacc[tid]);
}

// ---------------------------------------------------------------------------
__global__ __launch_bounds__(256) void k_hist(
    const unsigned* __restrict__ src, unsigned* __restrict__ counts, int shift)
{
  __shared__ unsigned lh[256];
  int tid = threadIdx.x;
  int seg = blockIdx.x >> 7;
  int blk = blockIdx.x & (RB - 1);
  lh[tid] = 0;
  __syncthreads();
  const unsigned* s = src + (size_t)seg * P_ + blk * RBI;
  for (int i = 0; i < 16; ++i) {
    unsigned d = (s[i * 256 + tid] >> shift) & 255u;
    atomicAdd(&lh[d], 1u);
  }
  __syncthreads();
  counts[((size_t)seg * 256 + tid) * RB + blk] = lh[tid];
}

__global__ __launch_bounds__(1024) void k_scan(unsigned* __restrict__ counts)
{
  __shared__ unsigned ps[1024];
  int tid = threadIdx.x;
  int seg = blockIdx.x;
  unsigned* c = counts + (size_t)seg * 256 * RB;  // 32768 entries, digit-major
  unsigned v[32];
  unsigned tot = 0;
  int base = tid * 32;
  #pragma unroll
  for (int i = 0; i < 32; ++i) { v[i] = c[base + i]; tot += v[i]; }
  ps[tid] = tot;
  __syncthreads();
  for (int off = 1; off < 1024; off <<= 1) {
    unsigned add = (tid >= off) ? ps[tid - off] : 0u;
    __syncthreads();
    ps[tid] += add;
    __syncthreads();
  }
  unsigned run = (unsigned)(seg * P_) + (ps[tid] - tot);  // exclusive + seg base
  #pragma unroll
  for (int i = 0; i < 32; ++i) { c[base + i] = run; run += v[i]; }
}

__global__ __launch_bounds__(256) void k_scatter(
    const unsigned* __restrict__ src, unsigned* __restrict__ dst,
    const unsigned* __restrict__ counts, int shift)
{
  __shared__ unsigned running[256];
  __shared__ unsigned wavecnt[8 * 256];
  int tid  = threadIdx.x;
  int lane = tid & 31, wave = tid >> 5;
  int seg  = blockIdx.x >> 7;
  int blk  = blockIdx.x & (RB - 1);
  running[tid] = counts[((size_t)seg * 256 + tid) * RB + blk];
  const unsigned* s = src + (size_t)seg * P_ + blk * RBI;

  for (int ch = 0; ch < 16; ++ch) {
    #pragma unroll
    for (int k = 0; k < 8; ++k) wavecnt[tid * 8 + k] = 0;
    __syncthreads();

    unsigned key = s[ch * 256 + tid];
    unsigned d   = (key >> shift) & 255u;
    // wave32 match-any via 8 ballots
    unsigned peers = 0xffffffffu;
    #pragma unroll
    for (int bb = 0; bb < 8; ++bb) {
      unsigned vote = __builtin_amdgcn_ballot_w32(((d >> bb) & 1u) != 0u);
      peers &= ((d >> bb) & 1u) ? vote : ~vote;
    }
    unsigned lower = (lane == 0) ? 0u : (peers & (0xffffffffu >> (32 - lane)));
    unsigned rank  = (unsigned)__popc(lower);
    if (rank == 0) wavecnt[wave * 256 + d] = (unsigned)__popc(peers);
    __syncthreads();

    unsigned cross = 0;
    for (int w = 0; w < wave; ++w) cross += wavecnt[w * 256 + d];
    dst[running[d] + cross + rank] = key;
    __syncthreads();

    unsigned tot = 0;
    #pragma unroll
    for (int w = 0; w < 8; ++w) tot += wavecnt[w * 256 + tid];
    running[tid] += tot;  // tid as digit
    __syncthreads();
  }
}

// ---------------------------------------------------------------------------
__global__ __launch_bounds__(1024) void k_lovasz(
    const unsigned* __restrict__ sorted, const float* __restrict__ stat,
    float* __restrict__ lov)
{
  __shared__ unsigned wtot[32];
  __shared__ float red[1024];
  int tid = threadIdx.x, lane = tid & 31, wave = tid >> 5;
  int seg = blockIdx.x;
  const unsigned* s = sorted + (size_t)seg * P_;
  float gts = stat[seg * 8 + 0];

  unsigned running = 0;
  float partial = 0.f;
  for (int ch = 0; ch < P_ / 1024; ++ch) {
    int i = ch * 1024 + tid;
    unsigned key = s[i];
    int lab = (int)(key & 1u);
    float ri = fmaxf(key_to_err(key), 0.f);
    float rnext = (i + 1 < P_) ? fmaxf(key_to_err(s[i + 1]), 0.f) : 0.f;

    unsigned bal  = __builtin_amdgcn_ballot_w32(lab != 0);
    unsigned incl = (unsigned)__popc(bal & (0xffffffffu >> (31 - lane)));
    __syncthreads();
    if (lane == 0) wtot[wave] = (unsigned)__popc(bal);
    __syncthreads();
    unsigned woff = 0, ctot = 0;
    for (int w = 0; w < 32; ++w) {
      unsigned x = wtot[w];
      ctot += x;
      if (w < wave) woff += x;
    }
    float cum = (float)(running + woff + incl);
    float U   = gts + (float)(i + 1) - cum;
    float jac = 1.f - (gts - cum) / U;
    partial += jac * (ri - rnext);
    running += ctot;
  }
  red[tid] = partial;
  __syncthreads();
  for (int off = 512; off > 0; off >>= 1) {
    if (tid < off) red[tid] += red[tid + off];
    __syncthreads();
  }
  if (tid == 0) lov[seg] = red[0];
}

// ---------------------------------------------------------------------------
__global__ void k_final(const float* __restrict__ stat,
                        const float* __restrict__ lov,
                        const float* __restrict__ fg,
                        const float* __restrict__ bgl,
                        float* __restrict__ out)
{
  if (threadIdx.x != 0 || blockIdx.x != 0) return;
  float lsum = 0.f, ssum = 0.f, dsum = 0.f;
  for (int b = 0; b < B_; ++b) {
    float npr = 0.f, lv = 0.f, sl = 0.f, fsum = 0.f;
    for (int n = 0; n < NI; ++n) {
      int seg = n * B_ + b;
      const float* o = stat + seg * 8;
      float pres = o[2], safe = o[1];
      npr  += pres;
      lv   += pres * lov[seg];
      sl   += pres * o[7];
      fsum += pres * fg[seg] / safe;
    }
    float np = fmaxf(npr, 1.f);
    lsum += lv / np;
    ssum += sl / np;
    dsum += (bgl[b] + fsum) / (1.f + npr);
  }
  out[0] = (lsum + ssum + dsum) * (1.f / (float)B_);
}

// ---------------------------------------------------------------------------
extern "C" void kernel_launch(void* const* d_in, const int* in_sizes, int n_in,
                              void* d_out, int out_size, void* d_ws, size_t ws_size,
                              hipStream_t stream)
{
  (void)in_sizes; (void)n_in; (void)out_size;
  const float* xvec  = (const float*)d_in[0];
  const float* xsig  = (const float*)d_in[1];
  const float* xseed = (const float*)d_in[2];
  const int*   t     = (const int*)d_in[3];
  float* out = (float*)d_out;

  char* ws = (char*)d_ws;
  float*    sums   = (float*)(ws + OFF_SUMS);
  float*    stat   = (float*)(ws + OFF_STAT);
  float*    bgl    = (float*)(ws + OFF_BG);
  float*    fg     = (float*)(ws + OFF_FG);
  float*    lov    = (float*)(ws + OFF_LOV);
  unsigned* counts = (unsigned*)(ws + OFF_CNTS);
  unsigned* itemsA = (unsigned*)(ws + OFF_ITEMA);
  unsigned* itemsB = (unsigned*)(ws + OFF_ITEMB);

  if (ws_size < WS_NEEDED) return;  // scratch too small: deterministic no-op

  k_zero  <<<4, 256, 0, stream>>>(sums, fg);
  k_reduce<<<dim3(P_ / 2048, B_), 256, 0, stream>>>(xvec, xsig, xseed, t, sums);
  k_stats <<<1, 32, 0, stream>>>(sums, stat, bgl);
  k_gauss <<<dim3(P_ / 2048, B_), 256, 0, stream>>>(xvec, xseed, t, stat, fg, itemsA);

  unsigned* src = itemsA;
  unsigned* dst = itemsB;
  for (int pass = 0; pass < 4; ++pass) {
    k_hist   <<<NSEG * RB, 256,  0, stream>>>(src, counts, pass * 8);
    k_scan   <<<NSEG,      1024, 0, stream>>>(counts);
    k_scatter<<<NSEG * RB, 256,  0, stream>>>(src, dst, counts, pass * 8);
    unsigned* tmp = src; src = dst; dst = tmp;
  }

  k_lovasz<<<NSEG, 1024, 0, stream>>>(src, stat, lov);
  k_final <<<1, 32, 0, stream>>>(stat, lov, fg, bgl, out);
}